// sknn_7799660610154
// MI455X (gfx1250) — compile-verified
//
#include <hip/hip_runtime.h>

#define QQ 4096
#define NN 50000
#define DD 5
#define NSPLIT 16
#define NTILES 3125   // NN / 16 exactly

typedef __attribute__((ext_vector_type(2))) float v2f;
typedef __attribute__((ext_vector_type(8))) float v8f;

// partial layout: [NSPLIT][QQ][3]  (sum, sum*locx, sum*locy)
__global__ __launch_bounds__(256) void nw_main_kernel(
    const float* __restrict__ rss,    // [QQ][5]
    const float* __restrict__ Radio,  // [NN][5]
    const float* __restrict__ Loc,    // [NN][2]
    const float* __restrict__ sigma,  // [1]
    float* __restrict__ partial) {
  const int lane = threadIdx.x & 31;
  const int wave = threadIdx.x >> 5;
  const int qt   = blockIdx.x * 8 + wave;   // 0..255 query tile
  const int chunk = blockIdx.y;             // 0..NSPLIT-1
  const int half = lane >> 4;               // 0 or 1
  const int mrow = lane & 15;

  // ---- build A (16x8, K padded): rows = queries of this tile ----
  const int qm = qt * 16 + mrow;
  const float* xp = rss + qm * DD;
  const float x0 = xp[0], x1 = xp[1], x2 = xp[2], x3 = xp[3], x4 = xp[4];
  const float x2m = x0*x0 + x1*x1 + x2*x2 + x3*x3 + x4*x4;

  // A0 covers K=0..3 (VGPR0: K=0|2, VGPR1: K=1|3), entries = -2*x
  v2f a0, a1;
  a0[0] = half ? (-2.0f * x2) : (-2.0f * x0);
  a0[1] = half ? (-2.0f * x3) : (-2.0f * x1);
  // A1 covers K=4..7: K=4 -> -2*x4, K=5 -> 1 (picks up r^2), K=6 -> x^2 (times 1), K=7 -> 0
  a1[0] = half ? x2m : (-2.0f * x4);
  a1[1] = half ? 0.0f : 1.0f;

  const float sg = sigma[0];
  // fold 1/(2*sigma^2) and log2(e) into one coefficient; use v_exp_f32 (base-2) directly
  const float coef2 = -1.44269504088896340736f / (2.0f * sg * sg);

  float acc_s[8], acc_x[8], acc_y[8];
#pragma unroll
  for (int j = 0; j < 8; ++j) { acc_s[j] = 0.0f; acc_x[j] = 0.0f; acc_y[j] = 0.0f; }

  const int t0 = (chunk * NTILES) / NSPLIT;
  const int t1 = ((chunk + 1) * NTILES) / NSPLIT;
  const float2* __restrict__ Loc2 = (const float2*)Loc;

  for (int t = t0; t < t1; ++t) {
    const int n = t * 16 + mrow;            // this lane's reference column
    const float* rp = Radio + n * DD;
    const float r0 = rp[0], r1 = rp[1], r2 = rp[2], r3 = rp[3], r4 = rp[4];
    const float rr = r0*r0 + r1*r1 + r2*r2 + r3*r3 + r4*r4;

    // B (8x16): lane = column n; VGPR0: K=k0, VGPR1: K=k0+1 (k0 = half*2)
    v2f b0, b1;
    b0[0] = half ? r2 : r0;
    b0[1] = half ? r3 : r1;
    // second WMMA: K=4 -> Radio[n][4], K=5 -> r^2, K=6 -> 1, K=7 -> 0
    b1[0] = half ? 1.0f : r4;
    b1[1] = half ? 0.0f : rr;

    const float2 lc = Loc2[n];

    // C accumulates directly to d2 = x^2 - 2 x.r + r^2
    v8f c = {};
    c = __builtin_amdgcn_wmma_f32_16x16x4_f32(false, a0, false, b0, (short)0, c, false, false);
    c = __builtin_amdgcn_wmma_f32_16x16x4_f32(false, a1, false, b1, (short)0, c, false, false);

#pragma unroll
    for (int j = 0; j < 8; ++j) {
      // single v_med3_f32 clamp to [0, +maxfloat] (avoids fmaxf canonicalize pair)
      float d2 = __builtin_amdgcn_fmed3f(c[j], 0.0f, 3.402823466e38f);
      float sim = __builtin_amdgcn_exp2f(d2 * coef2);   // one mul + v_exp_f32
      acc_s[j] += sim;
      acc_x[j] = fmaf(sim, lc.x, acc_x[j]);
      acc_y[j] = fmaf(sim, lc.y, acc_y[j]);
    }
  }

  // ---- reduce over the 16 reference columns held by each half-wave ----
#pragma unroll
  for (int j = 0; j < 8; ++j) {
    float s = acc_s[j], x = acc_x[j], y = acc_y[j];
#pragma unroll
    for (int m = 1; m < 16; m <<= 1) {
      s += __shfl_xor(s, m, 32);
      x += __shfl_xor(x, m, 32);
      y += __shfl_xor(y, m, 32);
    }
    if (mrow == 0) {
      const int q = qt * 16 + j + 8 * half; // lane 0 -> M=j, lane 16 -> M=j+8
      float* p = partial + ((size_t)chunk * QQ + q) * 3;
      p[0] = s; p[1] = x; p[2] = y;
    }
  }
}

__global__ __launch_bounds__(256) void nw_final_kernel(
    const float* __restrict__ partial, float* __restrict__ out) {
  const int q = blockIdx.x * blockDim.x + threadIdx.x;
  if (q >= QQ) return;
  float s = 0.0f, x = 0.0f, y = 0.0f;
#pragma unroll
  for (int c = 0; c < NSPLIT; ++c) {        // fixed order -> deterministic
    const float* p = partial + ((size_t)c * QQ + q) * 3;
    s += p[0]; x += p[1]; y += p[2];
  }
  out[q * 2 + 0] = x / s;
  out[q * 2 + 1] = y / s;
}

extern "C" void kernel_launch(void* const* d_in, const int* in_sizes, int n_in,
                              void* d_out, int out_size, void* d_ws, size_t ws_size,
                              hipStream_t stream) {
  const float* rss   = (const float*)d_in[0];
  const float* Radio = (const float*)d_in[1];
  const float* Loc   = (const float*)d_in[2];
  const float* sigma = (const float*)d_in[3];
  float* out = (float*)d_out;
  float* partial = (float*)d_ws;            // NSPLIT*QQ*3 floats = 768 KB

  dim3 grid(QQ / 16 / 8, NSPLIT);           // 32 x 16 blocks, 8 waves each
  nw_main_kernel<<<grid, 256, 0, stream>>>(rss, Radio, Loc, sigma, partial);
  nw_final_kernel<<<(QQ + 255) / 256, 256, 0, stream>>>(partial, out);
}